// DistEmb_60842506715846
// MI455X (gfx1250) — compile-verified
//
#include <hip/hip_runtime.h>
#include <hip/hip_bf16.h>

// DistEmb forward: out[i, :] = table[idx[i], :]   (row gather, EMB_DIM = 128 fp32)
//
// Pure HBM-bandwidth problem (~1.03 GB moved, ~45us floor at 23.3 TB/s).
// Mapping: one wave32 per row-slice -> 32 lanes x float4 = 512B = one full row
// per global_load_b128/global_store_b128 pair. 4 rows per wave for MLP.
// Indices are fetched 4-at-a-time with a single wave-uniform b128 load and
// moved to SGPRs with v_readfirstlane so table addressing is SGPR-base + VGPR
// offset. Output stores use non-temporal hints to keep L2 for the table
// (the gather has ~25% row reuse; streaming writes would otherwise evict it).

typedef float v4f __attribute__((ext_vector_type(4)));
typedef int   v4i __attribute__((ext_vector_type(4)));

#define EMB_DIM          128
#define ROWS_PER_WAVE    4
#define WAVES_PER_BLOCK  8
#define BLOCK_THREADS    (WAVES_PER_BLOCK * 32)
#define ROWS_PER_BLOCK   (WAVES_PER_BLOCK * ROWS_PER_WAVE)

__global__ __launch_bounds__(BLOCK_THREADS)
void DistEmb_gather_kernel(const float* __restrict__ table,
                           const int*   __restrict__ idx,
                           float*       __restrict__ out,
                           int n_rows)
{
    const int lane = threadIdx.x & 31;           // wave32
    const int wave = threadIdx.x >> 5;
    const long long rowBase =
        ((long long)blockIdx.x * WAVES_PER_BLOCK + wave) * ROWS_PER_WAVE;

    const int off = lane * 4;                    // float4 slot within the row

    if (rowBase + ROWS_PER_WAVE <= (long long)n_rows) {
        // One b128 load fetches all four row indices (wave-uniform address ->
        // broadcast). readfirstlane promotes them to SGPRs so each table base
        // is scalar; per-lane part is just `off`.
        v4i ids = *(const v4i*)(idx + rowBase);
        const size_t r0 = (size_t)__builtin_amdgcn_readfirstlane(ids.x) * EMB_DIM;
        const size_t r1 = (size_t)__builtin_amdgcn_readfirstlane(ids.y) * EMB_DIM;
        const size_t r2 = (size_t)__builtin_amdgcn_readfirstlane(ids.z) * EMB_DIM;
        const size_t r3 = (size_t)__builtin_amdgcn_readfirstlane(ids.w) * EMB_DIM;

        // 4 outstanding b128 gathers per wave before any store (hide HBM
        // latency on the random-read side).
        v4f d0 = *(const v4f*)(table + r0 + off);
        v4f d1 = *(const v4f*)(table + r1 + off);
        v4f d2 = *(const v4f*)(table + r2 + off);
        v4f d3 = *(const v4f*)(table + r3 + off);

        // Streaming output: non-temporal b128 stores (don't pollute the
        // 192MB L2, which we want to keep serving table-row reuse).
        float* o = out + (size_t)rowBase * EMB_DIM + off;
        __builtin_nontemporal_store(d0, (v4f*)(o + 0 * EMB_DIM));
        __builtin_nontemporal_store(d1, (v4f*)(o + 1 * EMB_DIM));
        __builtin_nontemporal_store(d2, (v4f*)(o + 2 * EMB_DIM));
        __builtin_nontemporal_store(d3, (v4f*)(o + 3 * EMB_DIM));
    } else {
        // Tail (not hit for n_rows = 1,048,576, kept for generality).
        for (int k = 0; k < ROWS_PER_WAVE; ++k) {
            const long long r = rowBase + k;
            if (r < (long long)n_rows) {
                const size_t src = (size_t)idx[r] * EMB_DIM;
                v4f d = *(const v4f*)(table + src + off);
                __builtin_nontemporal_store(
                    d, (v4f*)(out + (size_t)r * EMB_DIM + off));
            }
        }
    }
}

extern "C" void kernel_launch(void* const* d_in, const int* in_sizes, int n_in,
                              void* d_out, int out_size, void* d_ws, size_t ws_size,
                              hipStream_t stream)
{
    const float* table = (const float*)d_in[0];   // [NUM_NODES, 128] fp32
    const int*   idx   = (const int*)d_in[1];     // [NUM_IDS] int
    float*       out   = (float*)d_out;           // [NUM_IDS, 128] fp32

    const int n_rows = in_sizes[1];               // 1,048,576
    const int grid   = (n_rows + ROWS_PER_BLOCK - 1) / ROWS_PER_BLOCK;  // 32768

    DistEmb_gather_kernel<<<grid, BLOCK_THREADS, 0, stream>>>(table, idx, out, n_rows);
}